// Attention_15393162789165
// MI455X (gfx1250) — compile-verified
//
#include <hip/hip_runtime.h>
#include <math.h>

typedef float v2f __attribute__((ext_vector_type(2)));
typedef float v8f __attribute__((ext_vector_type(8)));

#define BB     8
#define SENC   256
#define SDEC   128
#define DIM    512   // D_ENC = D_DEC = UNITS = 512

__device__ __forceinline__ float dev_tanhf(float x) {
#if __has_builtin(__builtin_amdgcn_tanhf)
    return __builtin_amdgcn_tanhf(x);     // v_tanh_f32 on gfx1250
#else
    return tanhf(x);
#endif
}

// ---------------------------------------------------------------------------
// Y[M,512] = X[M,512] * W[512,512] + bias[512]
// One 16x16 output tile per wave via V_WMMA_F32_16X16X4_F32 (fp32-exact path).
// Block = 256 threads = 8 waves arranged 2(M) x 4(N): block tile = 32x64.
// Grid = (N/64, M/32).
// ---------------------------------------------------------------------------
__global__ __launch_bounds__(256) void proj_gemm_wmma(
    const float* __restrict__ X, const float* __restrict__ W,
    const float* __restrict__ bias, float* __restrict__ Y)
{
    const int K = 512, N = 512;
    const int lane = threadIdx.x & 31;
    const int wid  = threadIdx.x >> 5;
    const int half = lane >> 4;      // 0: K={0,1}/rows 0..7, 1: K={2,3}/rows 8..15
    const int lidx = lane & 15;
    const int tileM = blockIdx.y * 32 + (wid & 1) * 16;
    const int tileN = blockIdx.x * 64 + (wid >> 1) * 16;

#if __has_builtin(__builtin_amdgcn_wmma_f32_16x16x4_f32)
    v8f acc = {0.f, 0.f, 0.f, 0.f, 0.f, 0.f, 0.f, 0.f};
    const float* Arow = X + (size_t)(tileM + lidx) * K;   // A: row per lane (M=lidx)
    const float* Bcol = W + tileN + lidx;                 // B: col per lane (N=lidx)
    for (int k = 0; k < K; k += 4) {
        const int kk = k + 2 * half;
        v2f a, b;
        a.x = Arow[kk];                    // A VGPR0: K = 2*half
        a.y = Arow[kk + 1];                // A VGPR1: K = 2*half+1
        b.x = Bcol[(size_t)kk * N];        // B VGPR0: K = 2*half
        b.y = Bcol[(size_t)(kk + 1) * N];  // B VGPR1: K = 2*half+1
        acc = __builtin_amdgcn_wmma_f32_16x16x4_f32(
            /*neg_a=*/false, a, /*neg_b=*/false, b,
            /*c_mod=*/(short)0, acc, /*reuse_a=*/false, /*reuse_b=*/false);
    }
    const float bv = bias[tileN + lidx];
    #pragma unroll
    for (int r = 0; r < 8; ++r) {
        const int row = tileM + r + 8 * half;              // D: VGPR r -> M = r (+8 for hi lanes)
        Y[(size_t)row * N + tileN + lidx] = acc[r] + bv;
    }
#else
    // Scalar fallback (compiled only if the f32 WMMA builtin is absent)
    #pragma unroll
    for (int r = 0; r < 8; ++r) {
        const int row = tileM + r + 8 * half;
        const int col = tileN + lidx;
        float s = bias[col];
        for (int k = 0; k < K; ++k)
            s += X[(size_t)row * K + k] * W[(size_t)k * N + col];
        Y[(size_t)row * N + col] = s;
    }
#endif
}

// ---------------------------------------------------------------------------
// Fused Bahdanau scores + softmax + context.
// One block per (b,q); 256 threads; thread t owns encoder position e = t.
// ---------------------------------------------------------------------------
__global__ __launch_bounds__(256) void attn_fused(
    const float* __restrict__ enc,     // [B,SENC,DIM] raw encodings
    const float* __restrict__ encp,    // [B,SENC,DIM] projected + bias
    const float* __restrict__ decp,    // [B,SDEC,DIM] projected + bias
    const float* __restrict__ wscore,  // [DIM]
    float* __restrict__ out)           // [B,SDEC,DIM]
{
    __shared__ float dp[DIM];
    __shared__ float ws[DIM];
    __shared__ float wts[SENC];
    __shared__ float red[8];

    const int t  = threadIdx.x;
    const int bq = blockIdx.x;        // 0 .. B*SDEC-1
    const int b  = bq >> 7;           // / SDEC

    const float* dprow = decp + (size_t)bq * DIM;
    dp[t]       = dprow[t];
    dp[t + 256] = dprow[t + 256];
    ws[t]       = wscore[t];
    ws[t + 256] = wscore[t + 256];
    __syncthreads();

    // ---- score for e = t: sum_u ws[u] * tanh(dp[u] + encp[b,e,u]) ----
    const float* ep = encp + ((size_t)b * SENC + t) * DIM;
    float s = 0.f;
    for (int u = 0; u < DIM; u += 4) {
        const float4 e4 = *(const float4*)(ep + u);
        s += ws[u + 0] * dev_tanhf(dp[u + 0] + e4.x);
        s += ws[u + 1] * dev_tanhf(dp[u + 1] + e4.y);
        s += ws[u + 2] * dev_tanhf(dp[u + 2] + e4.z);
        s += ws[u + 3] * dev_tanhf(dp[u + 3] + e4.w);
    }

    // ---- block softmax over 256 encoder positions ----
    const int lane = t & 31, wid = t >> 5;
    float m = s;
    for (int off = 16; off > 0; off >>= 1) m = fmaxf(m, __shfl_xor(m, off, 32));
    if (lane == 0) red[wid] = m;
    __syncthreads();
    m = red[0];
    #pragma unroll
    for (int i = 1; i < 8; ++i) m = fmaxf(m, red[i]);
    const float ex = __expf(s - m);
    __syncthreads();                 // red[] reuse barrier
    float sum = ex;
    for (int off = 16; off > 0; off >>= 1) sum += __shfl_xor(sum, off, 32);
    if (lane == 0) red[wid] = sum;
    __syncthreads();
    sum = 0.f;
    #pragma unroll
    for (int i = 0; i < 8; ++i) sum += red[i];
    wts[t] = ex / sum;
    __syncthreads();

    // ---- context: out[b,q,d] = sum_e wts[e] * enc[b,e,d]; thread owns d=t, t+256 ----
    const float* eb = enc + (size_t)b * SENC * DIM;
    float c0 = 0.f, c1 = 0.f;
    for (int e = 0; e < SENC; ++e) {
        const float w = wts[e];
        c0 += w * eb[(size_t)e * DIM + t];
        c1 += w * eb[(size_t)e * DIM + t + 256];
    }
    out[(size_t)bq * DIM + t]       = c0;
    out[(size_t)bq * DIM + t + 256] = c1;
}

// ---------------------------------------------------------------------------
extern "C" void kernel_launch(void* const* d_in, const int* in_sizes, int n_in,
                              void* d_out, int out_size, void* d_ws, size_t ws_size,
                              hipStream_t stream) {
    const float* encodings = (const float*)d_in[0];  // [B,SENC,DIM]
    const float* decodings = (const float*)d_in[1];  // [B,SDEC,DIM]
    const float* W_enc     = (const float*)d_in[2];  // [DIM,DIM]
    const float* W_dec     = (const float*)d_in[3];  // [DIM,DIM]
    const float* W_score   = (const float*)d_in[4];  // [DIM,1]
    const float* bias_enc  = (const float*)d_in[5];  // [DIM]
    const float* bias_dec  = (const float*)d_in[6];  // [DIM]
    // bias_score (d_in[7]) cancels under softmax -> unused.

    float* encp = (float*)d_ws;                               // B*SENC*DIM floats (4 MB)
    float* decp = encp + (size_t)BB * SENC * DIM;             // B*SDEC*DIM floats (2 MB)

    const dim3 blk(256);
    proj_gemm_wmma<<<dim3(DIM / 64, (BB * SENC) / 32), blk, 0, stream>>>(
        encodings, W_enc, bias_enc, encp);
    proj_gemm_wmma<<<dim3(DIM / 64, (BB * SDEC) / 32), blk, 0, stream>>>(
        decodings, W_dec, bias_dec, decp);
    attn_fused<<<dim3(BB * SDEC), blk, 0, stream>>>(
        encodings, encp, decp, W_score, (float*)d_out);
}